// GraphConvolution_6794638262415
// MI455X (gfx1250) — compile-verified
//
#include <hip/hip_runtime.h>

typedef float v2f __attribute__((ext_vector_type(2)));
typedef float v8f __attribute__((ext_vector_type(8)));

#define IN_DIM  512
#define OUT_DIM 128
// Column-major W in LDS: wlds[col * CSTRIDE + k], CSTRIDE = 512 + 4 pad.
// A B-fragment (rows k+2h, k+2h+1 at col) is CONTIGUOUS -> one ds_load_b64
// into an aligned VGPR pair (direct WMMA B operand, no repacking movs).
// Banks: addr mod 64 = 4c + k + off (+2 per half) covers all 64 banks across
// the wave's 64 dwords -> conflict-free. Fragment base (n*16+m)*516+2h is
// loop-invariant; k*4 <= 2044 B fits the 16-bit DS offset -> 8 persistent
// base regs, zero address VALU in the k-loop.
#define CSTRIDE 516

// ---------------------------------------------------------------------------
// Phase 1: xw = x @ W via V_WMMA_F32_16X16X4_F32.
// One wave computes a 32-row x 128-col strip (2 M-tiles x 8 N-tiles),
// double-buffering A/B fragments across k-steps. 264 KB LDS => one 8-wave
// block per WGP (2 waves/SIMD); launch_bounds unlocks the VGPR budget.
// ---------------------------------------------------------------------------
__global__ void
__launch_bounds__(256, 2)
gemm_wmma_f32_kernel(const float* __restrict__ x,
                     const float* __restrict__ w,
                     float* __restrict__ xw,
                     int n_nodes, int n_pairs) {
    extern __shared__ float wlds[];

    // Stage W[512][128] column-major (coalesced reads; one-time).
    for (int idx = threadIdx.x; idx < IN_DIM * OUT_DIM; idx += blockDim.x) {
        int k   = idx >> 7;
        int col = idx & (OUT_DIM - 1);
        wlds[col * CSTRIDE + k] = w[idx];
    }
    __syncthreads();

    const int wave = threadIdx.x >> 5;
    const int lane = threadIdx.x & 31;
    const int pair = blockIdx.x * (blockDim.x >> 5) + wave;
    if (pair >= n_pairs) return;            // wave-uniform: EXEC all-1s for WMMA

    const int m0   = pair * 32;
    const int m    = lane & 15;
    const int half = lane >> 4;             // A/B fragment half (K 0,1 vs 2,3)

    int r0 = m0 + m;          if (r0 >= n_nodes) r0 = n_nodes - 1;
    int r1 = m0 + 16 + m;     if (r1 >= n_nodes) r1 = n_nodes - 1;
    const float* __restrict__ aptr0 = x + (size_t)r0 * IN_DIM + half * 2;
    const float* __restrict__ aptr1 = x + (size_t)r1 * IN_DIM + half * 2;

    // 8 loop-invariant per-lane B fragment bases (one per N-tile).
    const float* __restrict__ bptr[8];
#pragma unroll
    for (int n = 0; n < 8; ++n)
        bptr[n] = wlds + (n * 16 + m) * CSTRIDE + half * 2;

    v8f acc0[8] = {};
    v8f acc1[8] = {};

    // Prologue: fragments for k = 0.
    v2f a0 = *(const v2f*)(aptr0);
    v2f a1 = *(const v2f*)(aptr1);
    v2f b[8];
#pragma unroll
    for (int n = 0; n < 8; ++n)
        b[n] = *(const v2f*)(bptr[n]);      // ds_load_b64

#pragma unroll 2
    for (int k = 0; k < IN_DIM - 4; k += 4) {
        // Issue next-step loads first (independent of current fragments).
        v2f a0n = *(const v2f*)(aptr0 + k + 4);
        v2f a1n = *(const v2f*)(aptr1 + k + 4);
        v2f bn[8];
#pragma unroll
        for (int n = 0; n < 8; ++n)
            bn[n] = *(const v2f*)(bptr[n] + k + 4);   // ds_load_b64, imm offset
        // 16 back-to-back WMMAs on the current fragments (B reused 2x).
#pragma unroll
        for (int n = 0; n < 8; ++n)
            acc0[n] = __builtin_amdgcn_wmma_f32_16x16x4_f32(
                false, a0, false, b[n], (short)0, acc0[n], false, false);
#pragma unroll
        for (int n = 0; n < 8; ++n)
            acc1[n] = __builtin_amdgcn_wmma_f32_16x16x4_f32(
                false, a1, false, b[n], (short)0, acc1[n], false, false);
        a0 = a0n;
        a1 = a1n;
#pragma unroll
        for (int n = 0; n < 8; ++n) b[n] = bn[n];
    }
    // Epilogue: last k-step.
#pragma unroll
    for (int n = 0; n < 8; ++n)
        acc0[n] = __builtin_amdgcn_wmma_f32_16x16x4_f32(
            false, a0, false, b[n], (short)0, acc0[n], false, false);
#pragma unroll
    for (int n = 0; n < 8; ++n)
        acc1[n] = __builtin_amdgcn_wmma_f32_16x16x4_f32(
            false, a1, false, b[n], (short)0, acc1[n], false, false);

    // D layout: lanes 0-15 -> M=v, lanes 16-31 -> M=8+v; N = lane&15.
#pragma unroll
    for (int n = 0; n < 8; ++n) {
        const int col = n * 16 + m;
#pragma unroll
        for (int v = 0; v < 8; ++v) {
            const int ra = m0 + half * 8 + v;
            const int rb = m0 + 16 + half * 8 + v;
            if (ra < n_nodes) xw[(size_t)ra * OUT_DIM + col] = acc0[n][v];
            if (rb < n_nodes) xw[(size_t)rb * OUT_DIM + col] = acc1[n][v];
        }
    }
}

// ---------------------------------------------------------------------------
// Phase 2: COO scatter  out[r] += val * xw[c]  (L2-resident gather + atomics).
// One wave covers all 128 columns (float4/lane). Edge metadata loaded
// coalesced (one edge per lane), broadcast via __shfl.
// ---------------------------------------------------------------------------
__global__ void spmm_scatter_kernel(const float* __restrict__ xw,
                                    const int*   __restrict__ erow,
                                    const int*   __restrict__ ecol,
                                    const float* __restrict__ eval,
                                    float* __restrict__ out, int n_edges) {
    const int lane  = threadIdx.x & 31;
    const int wave  = (blockIdx.x * blockDim.x + threadIdx.x) >> 5;
    const int nwave = (gridDim.x * blockDim.x) >> 5;

    for (int e0 = wave * 32; e0 < n_edges; e0 += nwave * 32) {
        const int e = e0 + lane;
        int r = 0, c = 0;
        float v = 0.0f;
        if (e < n_edges) {
            r = erow[e];
            c = ecol[e];
            v = eval[e];
        }
        const int ep = e + nwave * 32;       // prefetch next metadata chunk
        if (ep < n_edges) {
            __builtin_prefetch(&erow[ep], 0, 0);
            __builtin_prefetch(&ecol[ep], 0, 0);
            __builtin_prefetch(&eval[ep], 0, 0);
        }

        int cnt = n_edges - e0;
        if (cnt >= 32) {
            // Fast path: unroll so gathers batch ahead of the atomics.
#pragma unroll 4
            for (int i = 0; i < 32; ++i) {
                const int   rr = __shfl(r, i, 32);
                const int   cc = __shfl(c, i, 32);
                const float vv = __shfl(v, i, 32);
                const float4 s =
                    ((const float4*)(xw + (size_t)cc * OUT_DIM))[lane];
                float* dst = out + (size_t)rr * OUT_DIM + lane * 4;
                atomicAdd(dst + 0, s.x * vv);
                atomicAdd(dst + 1, s.y * vv);
                atomicAdd(dst + 2, s.z * vv);
                atomicAdd(dst + 3, s.w * vv);
            }
        } else {
            for (int i = 0; i < cnt; ++i) {
                const int   rr = __shfl(r, i, 32);
                const int   cc = __shfl(c, i, 32);
                const float vv = __shfl(v, i, 32);
                const float4 s =
                    ((const float4*)(xw + (size_t)cc * OUT_DIM))[lane];
                float* dst = out + (size_t)rr * OUT_DIM + lane * 4;
                atomicAdd(dst + 0, s.x * vv);
                atomicAdd(dst + 1, s.y * vv);
                atomicAdd(dst + 2, s.z * vv);
                atomicAdd(dst + 3, s.w * vv);
            }
        }
    }
}

// ---------------------------------------------------------------------------
// Phase 3: ReLU in place.
// ---------------------------------------------------------------------------
__global__ void relu_kernel(float* __restrict__ out, int n4) {
    const int i = blockIdx.x * blockDim.x + threadIdx.x;
    if (i < n4) {
        float4* p = (float4*)out;
        float4 v = p[i];
        v.x = fmaxf(v.x, 0.0f);
        v.y = fmaxf(v.y, 0.0f);
        v.z = fmaxf(v.z, 0.0f);
        v.w = fmaxf(v.w, 0.0f);
        p[i] = v;
    }
}

extern "C" void kernel_launch(void* const* d_in, const int* in_sizes, int n_in,
                              void* d_out, int out_size, void* d_ws, size_t ws_size,
                              hipStream_t stream) {
    const float* x    = (const float*)d_in[0];
    const float* w    = (const float*)d_in[1];
    const int*   erow = (const int*)d_in[2];
    const int*   ecol = (const int*)d_in[3];
    const float* eval = (const float*)d_in[4];
    float*       out  = (float*)d_out;
    float*       xw   = (float*)d_ws;    // [n_nodes * OUT_DIM] fp32 scratch

    const int n_nodes = in_sizes[0] / IN_DIM;
    const int n_edges = in_sizes[2];

    hipMemsetAsync(d_out, 0, (size_t)out_size * sizeof(float), stream);

    // GEMM: each wave does 32 rows; 8 waves per 256-thread block.
    const int n_tiles  = (n_nodes + 15) / 16;
    const int n_pairs  = (n_tiles + 1) / 2;
    const int tpb      = 256;
    const int waves_pb = tpb / 32;
    const int gblocks  = (n_pairs + waves_pb - 1) / waves_pb;
    const size_t lds_bytes = (size_t)OUT_DIM * CSTRIDE * sizeof(float); // 264 KB
    gemm_wmma_f32_kernel<<<gblocks, tpb, lds_bytes, stream>>>(x, w, xw,
                                                              n_nodes, n_pairs);

    // Scatter: 4096 waves grid-striding over 32-edge chunks.
    spmm_scatter_kernel<<<512, 256, 0, stream>>>(xw, erow, ecol, eval, out,
                                                 n_edges);

    // ReLU.
    const int n4 = out_size / 4;
    relu_kernel<<<(n4 + 255) / 256, 256, 0, stream>>>(out, n4);
}